// AlphaCutoffFilter_4870492914167
// MI455X (gfx1250) — compile-verified
//
#include <hip/hip_runtime.h>
#include <math.h>

// ---------------------------------------------------------------------------
// EMA (1st-order IIR) scan over T for x[B=32, T=8192, C=128], fp32.
//   y_0 = x_0 ; y_t = a_c*y_{t-1} + (1-a_c)*x_t
// Bandwidth-bound (256 MiB @ 23.3 TB/s ~= 11.5us). Strategy:
//   K1: per 256-step chunk, zero-init end-state partials (one coalesced pass).
//   K2: chunked scan. Per wave: one channel x 256 steps as Y = S*X using
//       4 chained V_WMMA_F32_16X16X4_F32 (S = 16x16 lower-tri decay matrix,
//       X columns = 16 sub-blocks of 16 steps), then carry fix-ups.
//       x tile staged to LDS via global_load_async_to_lds_b32 (ASYNCcnt),
//       output transposed via LDS, stored non-temporal (preserve L2 for x).
// The t=0 boundary is absorbed by seeding y_{-1} = x[b,0,c].
// Decay powers: branch-free square-and-multiply computed UNconditionally,
// with a single final select per coefficient -> straight-line WMMA block.
// ---------------------------------------------------------------------------

typedef __attribute__((ext_vector_type(2))) float v2f;
typedef __attribute__((ext_vector_type(8))) float v8f;

#define T_LEN   8192
#define C_LEN   128
#define CHUNK   256                  // timesteps per chunk = 16 sub-blocks x 16
#define NCHUNK  (T_LEN / CHUNK)      // 32
#define CPB     32                   // channels per block in kernel 2
#define NSUB    16
#define LDS_STRIDE 33                // 32 channels + 1 pad float (bank spread)
#define SMEM_FLOATS (CHUNK * LDS_STRIDE + 2 * CPB * NSUB)

__device__ __forceinline__ float alpha_of(float lfc) {
  float fc = __expf(lfc);
  fc = fminf(fmaxf(fc, 1e-4f), 0.5f);
  return 1.0f - __expf(-6.28318530717958647692f * fc);
}

// a^(e & 15), branch-free; always evaluated (callers mask the result).
__device__ __forceinline__ float pow4bit(float a, int e) {
  float r = 1.0f, b = a;
  r = (e & 1) ? r * b : r; b *= b;
  r = (e & 2) ? r * b : r; b *= b;
  r = (e & 4) ? r * b : r; b *= b;
  r = (e & 8) ? r * b : r;
  return r;
}

// ---------------- Kernel 1: zero-init chunk partials -----------------------
__global__ __launch_bounds__(C_LEN)
void partials_kernel(const float* __restrict__ x,
                     const float* __restrict__ log_fc,
                     float* __restrict__ partials) {
  const int c = threadIdx.x;      // 0..127, contiguous -> coalesced
  const int b = blockIdx.x;
  const int k = blockIdx.y;
  const float a   = alpha_of(log_fc[c]);
  const float oma = 1.0f - a;
  const float* p = x + ((size_t)b * T_LEN + (size_t)k * CHUNK) * C_LEN + c;
  float s = 0.0f;
#pragma unroll 8
  for (int t = 0; t < CHUNK; ++t)
    s = fmaf(a, s, oma * p[(size_t)t * C_LEN]);
  partials[((size_t)b * NCHUNK + k) * C_LEN + c] = s;
}

// ---------------- Kernel 2: WMMA chunked scan + carry ----------------------
__global__ __launch_bounds__(1024)
void scan_kernel(const float* __restrict__ x,
                 const float* __restrict__ log_fc,
                 const float* __restrict__ partials,
                 float* __restrict__ y) {
  extern __shared__ float smem[];
  float* lds_x = smem;                          // [CHUNK][LDS_STRIDE]
  float* ends  = smem + CHUNK * LDS_STRIDE;     // [CPB][NSUB] sub-block ends
  float* ccar  = ends + CPB * NSUB;             // [CPB][NSUB] sub-block carries

  const int tid  = threadIdx.x;
  const int wave = tid >> 5;        // 0..31 == channel-within-group
  const int lane = tid & 31;
  const int b = blockIdx.x;
  const int k = blockIdx.y;
  const int g = blockIdx.z;
  const int c = g * CPB + wave;
  const size_t t0 = (size_t)k * CHUNK;

  // ---- Phase A: async copy x[b, t0:t0+256, g*32:(g+1)*32] into LDS ----
  {
    const int ch = tid & (CPB - 1);
    const int r0 = tid >> 5;                      // rows r0, r0+32, ...
    const float* gbase =
        x + ((size_t)b * T_LEN + t0) * C_LEN + (size_t)g * CPB + ch;
#pragma unroll
    for (int it = 0; it < CHUNK / 32; ++it) {
      const int r = r0 + 32 * it;
      const unsigned lds_off = (unsigned)((r * LDS_STRIDE + ch) * 4);
      const unsigned long long gaddr =
          (unsigned long long)(const void*)(gbase + (size_t)r * C_LEN);
      asm volatile("global_load_async_to_lds_b32 %0, %1, off"
                   :: "v"(lds_off), "v"(gaddr) : "memory");
    }
    asm volatile("s_wait_asynccnt 0x0" ::: "memory");
  }
  __syncthreads();

  const float a   = alpha_of(log_fc[c]);
  const float oma = 1.0f - a;
  const int h = lane >> 4;          // lane half
  const int m = lane & 15;          // A row index, B/D column index

  // ---- Phase B: local zero-init scan of 16 sub-blocks: D = S * X ----
  // S[t,s] = (s<=t) ? (1-a)*a^(t-s) : 0   (16x16, split into 4 16x4 slices)
  // X[s,j] = x_tile[16*j + s]             (column j = sub-block j)
  v8f d = {};
#pragma unroll
  for (int ks = 0; ks < 4; ++ks) {
    const int s0 = 4 * ks + 2 * h;  // per-lane K indices: s0, s0+1
    const int ex = m - s0;
    const int ey = ex - 1;
    // Unconditional pow chains (squarings CSE across all slices), then one
    // cheap select each -> v_cndmask, no exec-mask branching near WMMA.
    const float px = oma * pow4bit(a, ex & 15);
    const float py = oma * pow4bit(a, ey & 15);
    v2f A;
    A.x = (ex >= 0) ? px : 0.0f;
    A.y = (ey >= 0) ? py : 0.0f;
    v2f Bv;
    Bv.x = lds_x[(m * 16 + s0    ) * LDS_STRIDE + wave];
    Bv.y = lds_x[(m * 16 + s0 + 1) * LDS_STRIDE + wave];
    d = __builtin_amdgcn_wmma_f32_16x16x4_f32(
        /*neg_a=*/false, A, /*neg_b=*/false, Bv,
        /*c_mod=*/(short)0, d, /*reuse_a=*/false, /*reuse_b=*/false);
  }
  __syncthreads();                  // all lds_x tile reads complete

  // ---- Phase C: carries. d[7], lanes 16..31 hold Y[15, j] (sub-block ends)
  if (h == 1) ends[wave * NSUB + m] = d[7];
  __syncthreads();
  if (lane == 0) {
    // constant-exponent decay factors: a^16 and a^256 (explicit squarings)
    const float a2 = a * a, a4 = a2 * a2, a8 = a4 * a4, a16 = a8 * a8;
    const float a32 = a16 * a16, a64 = a32 * a32, a128 = a64 * a64;
    const float d256 = a128 * a128;
    // chunk-incoming state: y_{-1} = x[b,0,c], then fold chunks 0..k-1
    float Cin = x[(size_t)b * T_LEN * C_LEN + c];
    for (int j = 0; j < k; ++j)
      Cin = fmaf(d256, Cin, partials[((size_t)b * NCHUNK + j) * C_LEN + c]);
    // sub-block carry scan (factor a^16)
    float cv = Cin;
#pragma unroll
    for (int j = 0; j < NSUB; ++j) {
      ccar[wave * NSUB + j] = cv;
      cv = fmaf(a16, cv, ends[wave * NSUB + j]);
    }
  }
  __syncthreads();

  // ---- Phase D: apply carry: Y[t, j] += a^(t+1) * cc_j ; t = v + 8h ----
  {
    const float cv = ccar[wave * NSUB + m];
    const float a2 = a * a, a4 = a2 * a2, a8 = a4 * a4;
    float pw = a * (h ? a8 : 1.0f);         // a^(8h+1), branch-free
#pragma unroll
    for (int v = 0; v < 8; ++v) {
      d[v] = fmaf(pw, cv, d[v]);
      pw *= a;
    }
  }

  // ---- Phase E: transpose result back through LDS (column = wave) ----
#pragma unroll
  for (int v = 0; v < 8; ++v) {
    const int t = m * 16 + v + 8 * h;     // global row within chunk
    lds_x[t * LDS_STRIDE + wave] = d[v];
  }
  __syncthreads();

  // ---- Phase F: coalesced non-temporal store (keep L2 for x) ----
  {
    const int ch = tid & (CPB - 1);
    const int r0 = tid >> 5;
    float* obase =
        y + ((size_t)b * T_LEN + t0) * C_LEN + (size_t)g * CPB + ch;
#pragma unroll
    for (int it = 0; it < CHUNK / 32; ++it) {
      const int r = r0 + 32 * it;
      __builtin_nontemporal_store(lds_x[r * LDS_STRIDE + ch],
                                  obase + (size_t)r * C_LEN);
    }
  }
}

// ---------------------------------------------------------------------------
extern "C" void kernel_launch(void* const* d_in, const int* in_sizes, int n_in,
                              void* d_out, int out_size, void* d_ws, size_t ws_size,
                              hipStream_t stream) {
  (void)n_in; (void)out_size; (void)ws_size;
  const float* x      = (const float*)d_in[0];
  const float* log_fc = (const float*)d_in[1];
  float* yout         = (float*)d_out;
  float* partials     = (float*)d_ws;   // B * NCHUNK * C floats (512 KB)

  const int B = in_sizes[0] / (T_LEN * C_LEN);

  partials_kernel<<<dim3(B, NCHUNK), C_LEN, 0, stream>>>(x, log_fc, partials);
  scan_kernel<<<dim3(B, NCHUNK, C_LEN / CPB), 1024,
                SMEM_FLOATS * sizeof(float), stream>>>(x, log_fc, partials, yout);
}